// Generator_58969900974169
// MI455X (gfx1250) — compile-verified
//
#include <hip/hip_runtime.h>

typedef __attribute__((ext_vector_type(2))) float v2f;
typedef __attribute__((ext_vector_type(8))) float v8f;

#define LENGTH 49152
#define FDIM   1024
#define NTILES (LENGTH / 16)     // 3072 16-row tiles
#define WPB    8                 // waves per block (256 threads, wave32)
#define NCHUNK (LENGTH / 256)    // 192 scan chunks per batch
#define BATCH  2

// Scratch: Y[l][v][c] table (3 MB) + per-chunk partial sums. Fully overwritten
// every call -> deterministic across graph replays.
__device__ __align__(16) float g_Y[(size_t)LENGTH * 16];
__device__ float4 g_part[BATCH * NCHUNK];

// ---------------------------------------------------------------------------
// Pass 1: Y[l, v, c] = sum_f relu((v-3)*W_in[f] + pos_enc[l,f]) * W_final[f,c]
// One wave per 16-row tile; V_WMMA_F32_16X16X4_F32, K=1024 in steps of 4.
// A (16x4 f32, 2 VGPRs): lane<16 -> row=lane, K=khalf{0,1}; lane>=16 -> K={2,3}
// B (4x16 f32):          columns N>=4 are zero (masked via cmask, no EXEC divergence)
// C/D (16x16 f32, 8 VGPRs): vgpr r, lanes 0-15 -> M=r; lanes 16-31 -> M=8+r
// ---------------------------------------------------------------------------
__global__ __launch_bounds__(256) void build_y_wmma(
    const float* __restrict__ W_in, const float* __restrict__ pos_enc,
    const float* __restrict__ W_final) {
  const int lane  = threadIdx.x & 31;
  const int wave  = threadIdx.x >> 5;
  const int tile  = blockIdx.x * WPB + wave;           // 0..3071 (exact)
  const int m     = lane & 15;                          // A row / B,D column
  const int khalf = (lane >> 4) << 1;                   // 0 or 2
  const float* prow = pos_enc + (size_t)(tile * 16 + m) * FDIM + khalf;
  const int   cn    = (m < 4) ? m : 0;                  // live output channel
  const float cmask = (m < 4) ? 1.0f : 0.0f;            // zero-pad B cols 4..15

  v8f acc0 = {}, acc1 = {}, acc2 = {}, acc3 = {};       // v = 0..3 (s = v-3)

  for (int kk = 0; kk < FDIM; kk += 4) {
    v2f p = *(const v2f*)(prow + kk);                   // pos_enc pair for this lane
    v2f w = *(const v2f*)(W_in + kk + khalf);           // W_in pair
    v2f b;                                              // B = W_final chunk (padded)
    b.x = W_final[(kk + khalf) * 4 + cn] * cmask;
    b.y = W_final[(kk + khalf + 1) * 4 + cn] * cmask;

    v2f a;
    // v=0 -> s=-3
    a.x = fmaxf(fmaf(-3.0f, w.x, p.x), 0.0f);
    a.y = fmaxf(fmaf(-3.0f, w.y, p.y), 0.0f);
    acc0 = __builtin_amdgcn_wmma_f32_16x16x4_f32(false, a, false, b, (short)0, acc0, false, false);
    // v=1 -> s=-2
    a.x = fmaxf(fmaf(-2.0f, w.x, p.x), 0.0f);
    a.y = fmaxf(fmaf(-2.0f, w.y, p.y), 0.0f);
    acc1 = __builtin_amdgcn_wmma_f32_16x16x4_f32(false, a, false, b, (short)0, acc1, false, false);
    // v=2 -> s=-1
    a.x = fmaxf(p.x - w.x, 0.0f);
    a.y = fmaxf(p.y - w.y, 0.0f);
    acc2 = __builtin_amdgcn_wmma_f32_16x16x4_f32(false, a, false, b, (short)0, acc2, false, false);
    // v=3 -> s=0
    a.x = fmaxf(p.x, 0.0f);
    a.y = fmaxf(p.y, 0.0f);
    acc3 = __builtin_amdgcn_wmma_f32_16x16x4_f32(false, a, false, b, (short)0, acc3, false, false);
  }

  // Scatter D: lanes with N<4 hold live channels. vgpr r -> row (mbase + r).
  const int mbase = (lane >> 4) * 8;
  if (m < 4) {
    float* yb = g_Y + (size_t)(tile * 16 + mbase) * 16 + m;
#pragma unroll
    for (int r = 0; r < 8; ++r) {
      yb[r * 16 + 0]  = acc0[r];
      yb[r * 16 + 4]  = acc1[r];
      yb[r * 16 + 8]  = acc2[r];
      yb[r * 16 + 12] = acc3[r];
    }
  }
}

// ---------------------------------------------------------------------------
// Pass 2a: per-chunk float4 sums of the gathered sequence y[b,l,:] = Y[l, ex[b,l], :]
// ---------------------------------------------------------------------------
__global__ __launch_bounds__(256) void chunk_sums(const int* __restrict__ ex) {
  const int b = blockIdx.y, chunk = blockIdx.x, tid = threadIdx.x;
  const int l = chunk * 256 + tid;
  const int v = ex[b * LENGTH + l];
  float4 y = *(const float4*)(g_Y + (size_t)l * 16 + v * 4);
  __shared__ float4 sm[256];
  sm[tid] = y;
  __syncthreads();
  for (int s = 128; s > 0; s >>= 1) {
    if (tid < s) {
      sm[tid].x += sm[tid + s].x; sm[tid].y += sm[tid + s].y;
      sm[tid].z += sm[tid + s].z; sm[tid].w += sm[tid + s].w;
    }
    __syncthreads();
  }
  if (tid == 0) g_part[b * NCHUNK + chunk] = sm[0];
}

// ---------------------------------------------------------------------------
// Pass 2b: exclusive scan of the 192 chunk sums per batch (one block per batch)
// ---------------------------------------------------------------------------
__global__ __launch_bounds__(256) void scan_chunks() {
  const int b = blockIdx.x, tid = threadIdx.x;
  __shared__ float4 sm[256];
  float4 z = make_float4(0.f, 0.f, 0.f, 0.f);
  sm[tid] = (tid < NCHUNK) ? g_part[b * NCHUNK + tid] : z;
  __syncthreads();
  for (int off = 1; off < 256; off <<= 1) {
    float4 add = z;
    if (tid >= off) add = sm[tid - off];
    __syncthreads();
    if (tid >= off) {
      sm[tid].x += add.x; sm[tid].y += add.y; sm[tid].z += add.z; sm[tid].w += add.w;
    }
    __syncthreads();
  }
  if (tid < NCHUNK) g_part[b * NCHUNK + tid] = (tid == 0) ? z : sm[tid - 1];
}

// ---------------------------------------------------------------------------
// Pass 2c: local inclusive scan + chunk offset, scale by fp16(1/(l+1)), + b_final
// ---------------------------------------------------------------------------
__global__ __launch_bounds__(256) void scan_final(const int* __restrict__ ex,
                                                  const float* __restrict__ b_final,
                                                  float* __restrict__ out) {
  const int b = blockIdx.y, chunk = blockIdx.x, tid = threadIdx.x;
  const int l = chunk * 256 + tid;
  const int v = ex[b * LENGTH + l];
  float4 y = *(const float4*)(g_Y + (size_t)l * 16 + v * 4);
  __shared__ float4 sm[256];
  sm[tid] = y;
  __syncthreads();
  for (int off = 1; off < 256; off <<= 1) {
    float4 add = make_float4(0.f, 0.f, 0.f, 0.f);
    if (tid >= off) add = sm[tid - off];
    __syncthreads();
    if (tid >= off) {
      sm[tid].x += add.x; sm[tid].y += add.y; sm[tid].z += add.z; sm[tid].w += add.w;
    }
    __syncthreads();
  }
  float4 t   = sm[tid];
  float4 off0 = g_part[b * NCHUNK + chunk];
  // factor computed in fp16 then widened, matching reference astype(f16).astype(f32)
  const float fac = (float)(_Float16)(1.0f / (float)(l + 1));
  const float4 bf = *(const float4*)b_final;
  float4 o;
  o.x = (t.x + off0.x) * fac + bf.x;
  o.y = (t.y + off0.y) * fac + bf.y;
  o.z = (t.z + off0.z) * fac + bf.z;
  o.w = (t.w + off0.w) * fac + bf.w;
  *(float4*)(out + ((size_t)b * LENGTH + l) * 4) = o;
}

extern "C" void kernel_launch(void* const* d_in, const int* in_sizes, int n_in,
                              void* d_out, int out_size, void* d_ws, size_t ws_size,
                              hipStream_t stream) {
  const int*   example = (const int*)d_in[0];
  const float* W_in    = (const float*)d_in[1];
  const float* pos_enc = (const float*)d_in[2];
  const float* W_final = (const float*)d_in[3];
  const float* b_final = (const float*)d_in[4];
  float* out = (float*)d_out;

  build_y_wmma<<<NTILES / WPB, 256, 0, stream>>>(W_in, pos_enc, W_final);
  dim3 g(NCHUNK, BATCH);
  chunk_sums<<<g, 256, 0, stream>>>(example);
  scan_chunks<<<BATCH, 256, 0, stream>>>();
  scan_final<<<g, 256, 0, stream>>>(example, b_final, out);
}